// BandSplitModule_51049981280706
// MI455X (gfx1250) — compile-verified
//
#include <hip/hip_runtime.h>
#include <hip/hip_bf16.h>

// BandSplit for MI455X (gfx1250, wave32, WMMA).
// out[b, n, band, t] = ( LN(x_band[b,t,:]) @ W_band + bias_band )[n]
// fp32 end-to-end using V_WMMA_F32_16X16X4_F32.

#define NBANDS 41
#define KMAX   192      // max d_in = 190, padded to mult of 4

// Band start freq index and band width (freq bins), replicating the NumPy
// double-precision computation in create_freq_bands exactly.
__device__ __constant__ int c_bstart[NBANDS] = {
    0,4,9,13,18,22,27,32,36,41,
    46,57,69,80,92,103,115,127,138,150,161,173,
    185,208,231,254,278,301,324,347,
    371,417,464,510,557,603,650,696,
    743,835,928};
__device__ __constant__ int c_bflen[NBANDS] = {
    4,5,4,5,4,5,5,4,5,5,
    11,12,11,12,11,12,12,11,12,11,12,12,
    23,23,23,24,23,23,23,24,
    46,47,46,47,46,47,46,47,
    92,93,95};

struct BandParams {
    const float* gamma[NBANDS];
    const float* beta [NBANDS];
    const float* W    [NBANDS];   // (d_in, 256) row-major
    const float* bias [NBANDS];   // (256)
};

typedef __attribute__((ext_vector_type(2))) float v2f;
typedef __attribute__((ext_vector_type(8))) float v8f;

__launch_bounds__(256)
__global__ void bandsplit_wmma_kernel(const float* __restrict__ spec,
                                      BandParams P,
                                      float* __restrict__ out)
{
    const int band = blockIdx.y;
    const int b    = blockIdx.z;
    const int t0   = blockIdx.x << 4;          // 16 time steps per block
    const int s    = c_bstart[band];
    const int din  = c_bflen[band] * 2;        // features = freq*2 channels
    const int kpad = (din + 3) & ~3;           // pad K to multiple of 4

    __shared__ float s_xn [KMAX * 16];         // xn[k][t], t contiguous
    __shared__ float s_sum[256];
    __shared__ float s_sq [256];
    __shared__ float s_mu [16];
    __shared__ float s_rinv[16];

    const int tid = threadIdx.x;
    const int tl  = tid & 15;                  // t_local
    const int kp  = tid >> 4;                  // k-partial slot 0..15

    // ---- Stage x into LDS (coalesced along t) + per-row partial sums ----
    float psum = 0.f, psq = 0.f;
    for (int k = kp; k < kpad; k += 16) {
        float v = 0.f;
        if (k < din) {
            const int f = s + (k >> 1);
            const int c = k & 1;
            v = spec[((size_t)(b * 1024 + f) * 1024 + (t0 + tl)) * 2 + c];
        }
        s_xn[k * 16 + tl] = v;                 // padded K slots hold 0
        psum += v;
        psq  += v * v;
    }
    s_sum[tid] = psum;
    s_sq [tid] = psq;
    __syncthreads();

    // ---- LayerNorm stats per time step ----
    if (tid < 16) {
        float su = 0.f, sq = 0.f;
        #pragma unroll
        for (int p = 0; p < 16; ++p) {
            su += s_sum[p * 16 + tid];
            sq += s_sq [p * 16 + tid];
        }
        const float inv = 1.0f / (float)din;
        const float mu  = su * inv;
        const float var = sq * inv - mu * mu;  // population variance (ddof=0)
        s_mu  [tid] = mu;
        s_rinv[tid] = __frsqrt_rn(var + 1e-5f);
    }
    __syncthreads();

    // ---- Normalize in place: xn = (x-mu)*rsqrt(var+eps)*gamma + beta ----
    {
        const float* __restrict__ gamma = P.gamma[band];
        const float* __restrict__ beta  = P.beta [band];
        for (int idx = tid; idx < kpad * 16; idx += 256) {
            const int k = idx >> 4;
            const int t = idx & 15;
            if (k < din) {
                const float x = s_xn[idx];
                s_xn[idx] = (x - s_mu[t]) * s_rinv[t] * gamma[k] + beta[k];
            } // padded K rows stay 0 -> contribute 0 in WMMA
        }
    }
    __syncthreads();

    // ---- GEMM: D(16n x 16t) = A(W^T 16n x K) * B(xn^T K x 16t) ----
    // f32 16x16x4 WMMA layouts (ISA 7.12.2):
    //   A: lane holds row M=lane%16; VGPR0 -> K = kb + (lane/16)*2,  VGPR1 -> +1
    //   B: lane holds col N=lane%16; VGPR0 -> K = kb + (lane/16)*2,  VGPR1 -> +1
    //   D: VGPR r -> M = r + 8*(lane/16), N = lane%16
    const int wave  = tid >> 5;                // 0..7 -> covers 32 channels
    const int lane  = tid & 31;
    const int lhalf = lane >> 4;               // 0/1
    const int l16   = lane & 15;
    const int n0    = wave << 5;               // base output channel
    const float* __restrict__ Wm = P.W[band];

    v8f acc0 = {};
    v8f acc1 = {};

    // Full K steps: no clamping needed (din mod 4 is 0 or 2; full steps
    // touch only rows < (din & ~3) <= din).
    const int kfull = din & ~3;
    int ldsoff = (lhalf << 1) * 16 + l16;      // LDS index of B element k0
    int woff0  = (lhalf << 1) * 256 + n0 + l16;// W index of A element (k0, n)
    for (int kb = 0; kb < kfull; kb += 4) {
        v2f bm;                                // xn^T tile from LDS
        bm.x = s_xn[ldsoff];
        bm.y = s_xn[ldsoff + 16];
        v2f a0, a1;                            // W^T tiles (coalesced rows)
        a0.x = Wm[woff0];
        a0.y = Wm[woff0 + 256];
        a1.x = Wm[woff0 + 16];
        a1.y = Wm[woff0 + 256 + 16];
        acc0 = __builtin_amdgcn_wmma_f32_16x16x4_f32(
                   false, a0, false, bm, (short)0, acc0, false, false);
        acc1 = __builtin_amdgcn_wmma_f32_16x16x4_f32(
                   false, a1, false, bm, (short)0, acc1, false, false);
        ldsoff += 4 * 16;
        woff0  += 4 * 256;
    }

    // Tail step (only when din % 4 == 2): B rows >= din are zero in LDS,
    // clamp W row index so the (irrelevant) A values stay in bounds.
    if (kfull != kpad) {
        const int k0  = kfull + (lhalf << 1);
        const int kk0 = min(k0,     din - 1);
        const int kk1 = min(k0 + 1, din - 1);
        v2f bm;
        bm.x = s_xn[k0 * 16 + l16];
        bm.y = s_xn[k0 * 16 + 16 + l16];
        v2f a0, a1;
        a0.x = Wm[kk0 * 256 + n0 + l16];
        a0.y = Wm[kk1 * 256 + n0 + l16];
        a1.x = Wm[kk0 * 256 + n0 + 16 + l16];
        a1.y = Wm[kk1 * 256 + n0 + 16 + l16];
        acc0 = __builtin_amdgcn_wmma_f32_16x16x4_f32(
                   false, a0, false, bm, (short)0, acc0, false, false);
        acc1 = __builtin_amdgcn_wmma_f32_16x16x4_f32(
                   false, a1, false, bm, (short)0, acc1, false, false);
    }

    // ---- Bias + store: out[b, n, band, t], t along lanes => 64B segments ----
    const float* __restrict__ bias = P.bias[band];
    #pragma unroll
    for (int r = 0; r < 8; ++r) {
        const int na = n0 + r + (lhalf << 3);
        const int nb = na + 16;
        const float ya = acc0[r] + bias[na];
        const float yb = acc1[r] + bias[nb];
        out[((size_t)(b * 256 + na) * NBANDS + band) * 1024 + t0 + l16] = ya;
        out[((size_t)(b * 256 + nb) * NBANDS + band) * 1024 + t0 + l16] = yb;
    }
}

extern "C" void kernel_launch(void* const* d_in, const int* in_sizes, int n_in,
                              void* d_out, int out_size, void* d_ws, size_t ws_size,
                              hipStream_t stream) {
    const float* spec = (const float*)d_in[0];
    BandParams P;
    for (int i = 0; i < NBANDS; ++i) {
        P.gamma[i] = (const float*)d_in[1 + 4 * i + 0];
        P.beta [i] = (const float*)d_in[1 + 4 * i + 1];
        P.W    [i] = (const float*)d_in[1 + 4 * i + 2];
        P.bias [i] = (const float*)d_in[1 + 4 * i + 3];
    }
    dim3 grid(1024 / 16, NBANDS, 8);   // (t-tiles, bands, batch)
    bandsplit_wmma_kernel<<<grid, 256, 0, stream>>>(spec, P, (float*)d_out);
}